// HarmonNet_16853451669593
// MI455X (gfx1250) — compile-verified
//
#include <hip/hip_runtime.h>

typedef __attribute__((ext_vector_type(2))) float v2f;
typedef __attribute__((ext_vector_type(8))) float v8f;

#define DIM 10
#define PAD 16
#define LAMBDA 1.0f
#define BLOCK_THREADS 256
#define SAMPLES_PER_BLOCK 256
#define LSTRIDE 17

// ---------------------------------------------------------------------------
// Kernel 1: build Wsym = W + W^T, V = inv(Wsym - lambda*I) (Gauss-Jordan),
// write zero-padded 16x16 V, 16x16 Wsym, 16-elem b into workspace.
// ---------------------------------------------------------------------------
__global__ void harmon_setup(const float* __restrict__ W,
                             const float* __restrict__ b,
                             float* __restrict__ ws) {
    if (threadIdx.x != 0) return;
    float Ws[DIM][DIM];
    float aug[DIM][2 * DIM];
    for (int i = 0; i < DIM; ++i)
        for (int j = 0; j < DIM; ++j)
            Ws[i][j] = W[i * DIM + j] + W[j * DIM + i];
    for (int i = 0; i < DIM; ++i) {
        for (int j = 0; j < DIM; ++j) {
            aug[i][j]       = Ws[i][j] - (i == j ? LAMBDA : 0.0f);
            aug[i][DIM + j] = (i == j) ? 1.0f : 0.0f;
        }
    }
    for (int p = 0; p < DIM; ++p) {
        float inv = 1.0f / aug[p][p];
        for (int j = 0; j < 2 * DIM; ++j) aug[p][j] *= inv;
        for (int i = 0; i < DIM; ++i) {
            if (i == p) continue;
            float f = aug[i][p];
            for (int j = 0; j < 2 * DIM; ++j) aug[i][j] -= f * aug[p][j];
        }
    }
    float* Vp = ws;                 // [16][16]   B-matrix of h = y @ V
    float* Wp = ws + PAD * PAD;     // [16][16]   B-matrix of g = h @ Wsym
    float* bp = ws + 2 * PAD * PAD; // [16]
    for (int k = 0; k < PAD; ++k)
        for (int n = 0; n < PAD; ++n) {
            Vp[k * PAD + n] = (k < DIM && n < DIM) ? aug[k][DIM + n] : 0.0f;
            Wp[k * PAD + n] = (k < DIM && n < DIM) ? Ws[k][n] : 0.0f;
        }
    for (int n = 0; n < PAD; ++n) bp[n] = (n < DIM) ? b[n] : 0.0f;
}

// ---------------------------------------------------------------------------
// Wave-local LDS sync: all per-wave tile traffic (sX/sH/sG rows [32w,32w+32))
// stays within one wave, so a dscnt drain + scheduling barrier replaces the
// block-wide s_barrier — waves proceed independently.
// ---------------------------------------------------------------------------
__device__ __forceinline__ void wave_lds_sync() {
    asm volatile("s_wait_dscnt 0" ::: "memory");
    __builtin_amdgcn_wave_barrier();
}

// ---------------------------------------------------------------------------
// WMMA helpers (wave32, V_WMMA_F32_16X16X4_F32).
// A 16x4 f32 layout: lane m (0-15) row m; VGPR0=K0,VGPR1=K1; lanes 16-31: K2,K3.
// B 4x16 f32 layout: VGPR0 lanes0-15 = row K0 (N=lane), VGPR1 = K1;
//                    lanes 16-31 hold K2/K3 with N = lane-16.
// C/D 16x16: VGPR v, lanes0-15 -> (M=v, N=lane); lanes16-31 -> (M=v+8, N=lane-16).
// ---------------------------------------------------------------------------
__device__ __forceinline__ v8f tile_mm(const float (*src)[LSTRIDE], int base,
                                       const float* Bmat, int lm, int half) {
    v8f acc = {0.f, 0.f, 0.f, 0.f, 0.f, 0.f, 0.f, 0.f};
#pragma unroll
    for (int cch = 0; cch < 3; ++cch) {   // K = 0..11 covers DIM=10 (rest zero)
        const int k0 = 4 * cch + 2 * half;
        v2f A, Bv;
        A.x  = src[base + lm][k0];
        A.y  = src[base + lm][k0 + 1];
        Bv.x = Bmat[k0 * PAD + lm];
        Bv.y = Bmat[(k0 + 1) * PAD + lm];
        acc = __builtin_amdgcn_wmma_f32_16x16x4_f32(
            /*neg_a=*/false, A, /*neg_b=*/false, Bv,
            /*c_mod=*/(short)0, acc, /*reuse_a=*/false, /*reuse_b=*/false);
    }
    return acc;
}

__device__ __forceinline__ void store_tile(float (*dst)[LSTRIDE], int base,
                                           v8f acc, int lm, int half) {
#pragma unroll
    for (int v = 0; v < 8; ++v)
        dst[base + v + 8 * half][lm] = acc[v];
}

// ---------------------------------------------------------------------------
// Kernel 2: main. 256 threads = 8 waves; 256 samples per block.
// ---------------------------------------------------------------------------
__launch_bounds__(BLOCK_THREADS, 2)
__global__ void harmon_main(const int* __restrict__ samples,
                            const float* __restrict__ ent,
                            const float* __restrict__ rel,
                            const float* __restrict__ ws,
                            float* __restrict__ out, int total) {
    __shared__ float sV[PAD * PAD];
    __shared__ float sW[PAD * PAD];
    __shared__ float sb[PAD];
    __shared__ float sX[SAMPLES_PER_BLOCK][LSTRIDE];  // x
    __shared__ float sH[SAMPLES_PER_BLOCK][LSTRIDE];  // y, then h
    __shared__ float sG[SAMPLES_PER_BLOCK][LSTRIDE];  // g = h @ Wsym

    const int tid  = threadIdx.x;
    const int lane = tid & 31;
    const int wv   = tid >> 5;
    const int half = lane >> 4;
    const int lm   = lane & 15;

    // Load uniform matrices into LDS (block-wide, needs one real barrier).
    for (int i = tid; i < PAD * PAD; i += BLOCK_THREADS) {
        sV[i] = ws[i];
        sW[i] = ws[PAD * PAD + i];
    }
    if (tid < PAD) sb[tid] = ws[2 * PAD * PAD + tid];
    __syncthreads();

    // ---------------- Phase A: per-lane gather + circular correlation -------
    const int s  = blockIdx.x * SAMPLES_PER_BLOCK + tid;
    const int si = s < total ? s : total - 1;   // clamp, keep EXEC uniform
    const int e1 = samples[3 * si + 0];
    const int rl = samples[3 * si + 1];
    const int e2 = samples[3 * si + 2];

    // Row base addresses are idx*40 bytes -> always 8B aligned: use float2
    // (global_load_b64) gathers, 5 per row instead of 10 scalar loads.
    const float2* p1 = (const float2*)(ent + (size_t)e1 * DIM);
    const float2* p2 = (const float2*)(ent + (size_t)e2 * DIM);
    const float2* pr = (const float2*)(rel + (size_t)rl * DIM);

    float a[DIM], c[DIM], rr[DIM];
#pragma unroll
    for (int j = 0; j < DIM / 2; ++j) {
        const float2 va = p1[j];
        const float2 vc = p2[j];
        const float2 vr = pr[j];
        a[2 * j]      = va.x;  a[2 * j + 1]  = va.y;
        c[2 * j]      = vc.x;  c[2 * j + 1]  = vc.y;
        rr[2 * j]     = vr.x;  rr[2 * j + 1] = vr.y;
    }
#pragma unroll
    for (int k = 0; k < DIM; ++k) {
        float cc = 0.0f;
#pragma unroll
        for (int j = 0; j < DIM; ++j) cc += a[j] * c[(j + k) % DIM];
        const float x = rr[k] * cc;
        sX[tid][k] = x;
        sH[tid][k] = -0.5f * sb[k] + LAMBDA * x;   // y
    }
#pragma unroll
    for (int k = DIM; k < PAD; ++k) { sX[tid][k] = 0.0f; sH[tid][k] = 0.0f; }
    wave_lds_sync();   // wave-local: rows [32w,32w+32) only touched by wave w

    // ---------------- Phase B: WMMA matmuls (2 tiles of 16 rows per wave) ---
    const int base0 = 32 * wv;
    const int base1 = 32 * wv + 16;

    // h = y @ V
    v8f h0 = tile_mm(sH, base0, sV, lm, half);
    v8f h1 = tile_mm(sH, base1, sV, lm, half);
    wave_lds_sync();
    store_tile(sH, base0, h0, lm, half);   // overwrite y with h
    store_tile(sH, base1, h1, lm, half);
    wave_lds_sync();

    // g = h @ Wsym
    v8f g0 = tile_mm(sH, base0, sW, lm, half);
    v8f g1 = tile_mm(sH, base1, sW, lm, half);
    store_tile(sG, base0, g0, lm, half);
    store_tile(sG, base1, g1, lm, half);
    wave_lds_sync();

    // ---------------- Phase C: per-lane reductions --------------------------
    float quad = 0.0f, lin = 0.0f, nrm = 0.0f;
#pragma unroll
    for (int k = 0; k < DIM; ++k) {
        const float h = sH[tid][k];
        const float g = sG[tid][k];
        quad += g * h;
        lin  += h * sb[k];
        const float d = h - sX[tid][k];
        nrm  += d * d;
    }
    if (s < total) out[s] = quad + lin - LAMBDA * nrm;
}

// ---------------------------------------------------------------------------
extern "C" void kernel_launch(void* const* d_in, const int* in_sizes, int n_in,
                              void* d_out, int out_size, void* d_ws, size_t ws_size,
                              hipStream_t stream) {
    const int*   samples = (const int*)d_in[0];
    const float* ent     = (const float*)d_in[1];
    const float* rel     = (const float*)d_in[2];
    const float* W       = (const float*)d_in[3];
    const float* b       = (const float*)d_in[4];
    float*       out     = (float*)d_out;
    float*       ws      = (float*)d_ws;

    const int total  = in_sizes[0] / 3;                 // B*S samples
    const int blocks = (total + SAMPLES_PER_BLOCK - 1) / SAMPLES_PER_BLOCK;

    harmon_setup<<<1, 32, 0, stream>>>(W, b, ws);
    harmon_main<<<blocks, BLOCK_THREADS, 0, stream>>>(samples, ent, rel, ws, out, total);
}